// GATLayer_72387378806988
// MI455X (gfx1250) — compile-verified
//
#include <hip/hip_runtime.h>
#include <hip/hip_bf16.h>
#include <math.h>

#define N_NODES   8000
#define N_EDGES   64000
#define E_TOT     (N_EDGES + N_NODES)
#define HEADS     8
#define OUT_C     512
#define HC        (HEADS * OUT_C)   // 4096
#define IN_C      128
#define NEG_SLOPE 0.2f
#define GAT_EPS   1e-16f

typedef __attribute__((ext_vector_type(2))) float v2f;
typedef __attribute__((ext_vector_type(8))) float v8f;

// ---------------------------------------------------------------------------
// float atomic max via signed/unsigned integer atomics (works with -inf init)
// ---------------------------------------------------------------------------
__device__ __forceinline__ void atomic_max_f32(float* addr, float val) {
    if (val >= 0.0f) {
        atomicMax((int*)addr, __float_as_int(val));
    } else {
        atomicMin((unsigned int*)addr, __float_as_uint(val));
    }
}

// ---------------------------------------------------------------------------
// WMMA f32 GEMM: C[M,N] = A[M,K] @ B[K,N] (+ bias[N]); all row-major.
// One wave32 per 16x16 tile; K-loop of V_WMMA_F32_16X16X4_F32.
// M,N multiples of 16; K multiple of 4. EXEC is all-ones (no divergence).
// ---------------------------------------------------------------------------
__global__ __launch_bounds__(256) void wmma_gemm_f32(
    const float* __restrict__ A, const float* __restrict__ B,
    float* __restrict__ C, const float* __restrict__ bias,
    int N, int K, int tilesN) {

    const int wave  = blockIdx.x * (blockDim.x >> 5) + (threadIdx.x >> 5);
    const int tileM = wave / tilesN;
    const int tileN = wave % tilesN;
    const int lane  = threadIdx.x & 31;
    const int hl    = lane >> 4;       // 0: K pair {0,1}, 1: K pair {2,3}
    const int l     = lane & 15;       // row (A) / col (B,C) within tile

    const float* Ap = A + (size_t)(tileM * 16 + l) * K;   // row of A
    const float* Bp = B + (size_t)(tileN * 16 + l);       // col of B

    v8f acc = {};
    for (int k0 = 0; k0 < K; k0 += 4) {
        v2f a, b;
        // A 16x4 fragment: lanes 0-15 hold K=k0,k0+1; lanes 16-31 hold K=k0+2,k0+3
        a.x = Ap[k0 + 2 * hl];
        a.y = Ap[k0 + 2 * hl + 1];
        // B 4x16 fragment: mirrored striping across lane halves
        const float* bp = Bp + (size_t)(k0 + 2 * hl) * N;
        b.x = bp[0];
        b.y = bp[N];
        acc = __builtin_amdgcn_wmma_f32_16x16x4_f32(
            /*neg_a=*/false, a, /*neg_b=*/false, b,
            /*c_mod=*/(short)0, acc, /*reuse_a=*/false, /*reuse_b=*/false);
    }

    const float bv = bias ? bias[tileN * 16 + l] : 0.0f;
#pragma unroll
    for (int r = 0; r < 8; ++r) {
        // C layout: VGPR r -> row r (lanes 0-15) / row r+8 (lanes 16-31)
        C[(size_t)(tileM * 16 + r + hl * 8) * N + tileN * 16 + l] = acc[r] + bv;
    }
}

// ---------------------------------------------------------------------------
// Per-(node,head) attention logits: a_src/a_dst = <h[n,hh,:], att_*[hh,:]>
// One wave per (node,head) pair, shuffle reduction (wave32).
// ---------------------------------------------------------------------------
__global__ __launch_bounds__(256) void attn_logits(
    const float* __restrict__ h, const float* __restrict__ att_src,
    const float* __restrict__ att_dst, float* __restrict__ a_src,
    float* __restrict__ a_dst) {

    const int pair = blockIdx.x * (blockDim.x >> 5) + (threadIdx.x >> 5);
    if (pair >= N_NODES * HEADS) return;
    const int n = pair / HEADS, hh = pair % HEADS;
    const int lane = threadIdx.x & 31;

    const float* hp = h + (size_t)n * HC + hh * OUT_C;
    const float* as = att_src + hh * OUT_C;
    const float* ad = att_dst + hh * OUT_C;

    float s = 0.0f, d = 0.0f;
    for (int c = lane; c < OUT_C; c += 32) {
        const float v = hp[c];
        s += v * as[c];
        d += v * ad[c];
    }
#pragma unroll
    for (int off = 16; off > 0; off >>= 1) {
        s += __shfl_xor(s, off, 32);
        d += __shfl_xor(d, off, 32);
    }
    if (lane == 0) { a_src[pair] = s; a_dst[pair] = d; }
}

__global__ void init_neg_inf(float* __restrict__ p, int n) {
    const int i = blockIdx.x * blockDim.x + threadIdx.x;
    if (i < n) p[i] = -INFINITY;
}

__device__ __forceinline__ void edge_endpoints(const int* __restrict__ ei,
                                               int e, int* src, int* dst) {
    if (e < N_EDGES) { *src = ei[e]; *dst = ei[N_EDGES + e]; }
    else             { *src = e - N_EDGES; *dst = e - N_EDGES; }  // self loop
}

// segment_max over destinations
__global__ __launch_bounds__(256) void edge_max(
    const int* __restrict__ ei, const float* __restrict__ a_src,
    const float* __restrict__ a_dst, float* __restrict__ emax) {
    const int e = blockIdx.x * blockDim.x + threadIdx.x;
    if (e >= E_TOT) return;
    int src, dst; edge_endpoints(ei, e, &src, &dst);
#pragma unroll
    for (int hh = 0; hh < HEADS; ++hh) {
        float v = a_src[src * HEADS + hh] + a_dst[dst * HEADS + hh];
        v = (v > 0.0f) ? v : NEG_SLOPE * v;
        atomic_max_f32(&emax[dst * HEADS + hh], v);
    }
}

// ez = exp(e - emax[dst]); denom = segment_sum(ez)
__global__ __launch_bounds__(256) void edge_exp(
    const int* __restrict__ ei, const float* __restrict__ a_src,
    const float* __restrict__ a_dst, const float* __restrict__ emax,
    float* __restrict__ ez, float* __restrict__ denom) {
    const int e = blockIdx.x * blockDim.x + threadIdx.x;
    if (e >= E_TOT) return;
    int src, dst; edge_endpoints(ei, e, &src, &dst);
#pragma unroll
    for (int hh = 0; hh < HEADS; ++hh) {
        float v = a_src[src * HEADS + hh] + a_dst[dst * HEADS + hh];
        v = (v > 0.0f) ? v : NEG_SLOPE * v;
        const float z = __expf(v - emax[dst * HEADS + hh]);
        ez[(size_t)e * HEADS + hh] = z;
        atomicAdd(&denom[dst * HEADS + hh], z);
    }
}

// agg[dst] += alpha * h[src]  (one block per edge, coalesced over channels)
__global__ __launch_bounds__(256) void scatter_msg(
    const int* __restrict__ ei, const float* __restrict__ h,
    const float* __restrict__ ez, const float* __restrict__ denom,
    float* __restrict__ agg) {
    const int e = blockIdx.x;
    int src, dst; edge_endpoints(ei, e, &src, &dst);
    __shared__ float s_alpha[HEADS];
    if (threadIdx.x < HEADS)
        s_alpha[threadIdx.x] = ez[(size_t)e * HEADS + threadIdx.x] /
                               (denom[dst * HEADS + threadIdx.x] + GAT_EPS);
    __syncthreads();
    const float* hp = h + (size_t)src * HC;
    float* ap = agg + (size_t)dst * HC;
    for (int c = threadIdx.x; c < HC; c += blockDim.x)
        atomicAdd(&ap[c], hp[c] * s_alpha[c >> 9]);   // head = c / 512
}

// bias2[j] = b_lin[j] + sum_k b_gat[k] * W_lin[k, j]
__global__ __launch_bounds__(256) void fold_bias(
    const float* __restrict__ b_gat, const float* __restrict__ W_lin,
    const float* __restrict__ b_lin, float* __restrict__ bias2) {
    const int j = blockIdx.x * blockDim.x + threadIdx.x;
    if (j >= 512) return;
    float s = b_lin[j];
    for (int k = 0; k < HC; ++k) s += b_gat[k] * W_lin[(size_t)k * 512 + j];
    bias2[j] = s;
}

// ---------------------------------------------------------------------------
extern "C" void kernel_launch(void* const* d_in, const int* in_sizes, int n_in,
                              void* d_out, int out_size, void* d_ws, size_t ws_size,
                              hipStream_t stream) {
    (void)in_sizes; (void)n_in; (void)out_size; (void)ws_size;
    const float* x       = (const float*)d_in[0];
    const int*   ei      = (const int*)d_in[1];   // [2, N_EDGES] int32
    const float* W_gat   = (const float*)d_in[2];
    const float* b_gat   = (const float*)d_in[3];
    const float* att_src = (const float*)d_in[4];
    const float* att_dst = (const float*)d_in[5];
    const float* W_lin   = (const float*)d_in[6];
    const float* b_lin   = (const float*)d_in[7];
    float* out = (float*)d_out;

    char* ws = (char*)d_ws;
    size_t off = 0;
    auto alloc = [&](size_t nfloats) {
        float* p = (float*)(ws + off);
        off += ((nfloats * sizeof(float) + 255) / 256) * 256;
        return p;
    };
    float* h      = alloc((size_t)N_NODES * HC);     // 131 MB
    float* agg    = alloc((size_t)N_NODES * HC);     // 131 MB
    float* a_src  = alloc((size_t)N_NODES * HEADS);
    float* a_dst  = alloc((size_t)N_NODES * HEADS);
    float* emax   = alloc((size_t)N_NODES * HEADS);
    float* denom  = alloc((size_t)N_NODES * HEADS);
    float* ez     = alloc((size_t)E_TOT * HEADS);
    float* bias2  = alloc(512);

    hipMemsetAsync(agg, 0, (size_t)N_NODES * HC * sizeof(float), stream);
    hipMemsetAsync(denom, 0, (size_t)N_NODES * HEADS * sizeof(float), stream);
    init_neg_inf<<<(N_NODES * HEADS + 255) / 256, 256, 0, stream>>>(
        emax, N_NODES * HEADS);

    // GEMM1: h = x @ W_gat   (M=8000, K=128, N=4096) -> 500*256 tiles
    {
        const int tilesN = HC / 16;                      // 256
        const int waves  = (N_NODES / 16) * tilesN;      // 128000
        wmma_gemm_f32<<<waves / 8, 256, 0, stream>>>(x, W_gat, h, nullptr,
                                                     HC, IN_C, tilesN);
    }

    attn_logits<<<(N_NODES * HEADS) / 8, 256, 0, stream>>>(h, att_src, att_dst,
                                                           a_src, a_dst);
    edge_max<<<(E_TOT + 255) / 256, 256, 0, stream>>>(ei, a_src, a_dst, emax);
    edge_exp<<<(E_TOT + 255) / 256, 256, 0, stream>>>(ei, a_src, a_dst, emax,
                                                      ez, denom);
    scatter_msg<<<E_TOT, 256, 0, stream>>>(ei, h, ez, denom, agg);
    fold_bias<<<2, 256, 0, stream>>>(b_gat, W_lin, b_lin, bias2);

    // GEMM2: out = agg @ W_lin + (b_gat@W_lin + b_lin)  (M=8000, K=4096, N=512)
    {
        const int tilesN = 512 / 16;                     // 32
        const int waves  = (N_NODES / 16) * tilesN;      // 16000
        wmma_gemm_f32<<<waves / 8, 256, 0, stream>>>(agg, W_lin, out, bias2,
                                                     512, HC, tilesN);
    }
}